// GiGCNConv_32968168964747
// MI455X (gfx1250) — compile-verified
//
#include <hip/hip_runtime.h>

typedef __attribute__((ext_vector_type(2))) float v2f;
typedef __attribute__((ext_vector_type(8))) float v8f;

#define N_NODES 100000
#define N_EDGES 3200000
#define D 64

// --- 1) deg[n] = 1.0 (self-loop weight, improved=False) -----------------
__global__ __launch_bounds__(256) void k_init_deg(float* __restrict__ deg) {
    int n = blockIdx.x * blockDim.x + threadIdx.x;
    if (n < N_NODES) deg[n] = 1.0f;
}

// --- 2) deg[dst[e]] += ew[e] -------------------------------------------
__global__ __launch_bounds__(256) void k_accum_deg(const long long* __restrict__ dst,
                                                   const float* __restrict__ ew,
                                                   float* __restrict__ deg) {
    int e = blockIdx.x * blockDim.x + threadIdx.x;
    if (e < N_EDGES) unsafeAtomicAdd(&deg[dst[e]], ew[e]);
}

// --- 3) deg -> 1/deg (0 if deg==0) -------------------------------------
__global__ __launch_bounds__(256) void k_invert_deg(float* __restrict__ deg) {
    int n = blockIdx.x * blockDim.x + threadIdx.x;
    if (n < N_NODES) {
        float d = deg[n];
        deg[n] = d > 0.0f ? 1.0f / d : 0.0f;
    }
}

// --- 4) h = x @ W via V_WMMA_F32_16X16X4_F32 ---------------------------
// One wave per 16-row tile of x; 4 N-tiles x 16 K-steps of K=4 WMMAs.
// A frag (16x4 f32): lane L -> M = L%16, VGPR0/1 = x[M, k + (L<16?0:2) + {0,1}]
// B frag (4x16 f32): lane L -> N = L%16, VGPR0/1 = W[k + (L<16?0:2) + {0,1}, N]
// C/D (16x16 f32):  lane L -> N = L%16, VGPR r -> M = r + (L<16?0:8)
__global__ __launch_bounds__(256) void k_gemm(const float* __restrict__ x,
                                              const float* __restrict__ W,
                                              float* __restrict__ h) {
    const int lane  = threadIdx.x & 31;
    const int tile  = blockIdx.x * 8 + (threadIdx.x >> 5);   // 16-row tile
    if (tile * 16 >= N_NODES) return;                        // wave-uniform exit
    const int row0  = tile * 16;
    const int m     = lane & 15;
    const int khalf = (lane >> 4) * 2;                       // 0 or 2

    v8f acc0 = {}, acc1 = {}, acc2 = {}, acc3 = {};
    const float* xrow = x + (size_t)(row0 + m) * D;

#pragma unroll
    for (int k = 0; k < D; k += 4) {
        v2f a = *(const v2f*)(xrow + k + khalf);
        const float* wr0 = W + (size_t)(k + khalf) * D;
        const float* wr1 = wr0 + D;
        v2f b0; b0.x = wr0[ 0 + m]; b0.y = wr1[ 0 + m];
        v2f b1; b1.x = wr0[16 + m]; b1.y = wr1[16 + m];
        v2f b2; b2.x = wr0[32 + m]; b2.y = wr1[32 + m];
        v2f b3; b3.x = wr0[48 + m]; b3.y = wr1[48 + m];
        acc0 = __builtin_amdgcn_wmma_f32_16x16x4_f32(false, a, false, b0, (short)0, acc0, false, false);
        acc1 = __builtin_amdgcn_wmma_f32_16x16x4_f32(false, a, false, b1, (short)0, acc1, false, false);
        acc2 = __builtin_amdgcn_wmma_f32_16x16x4_f32(false, a, false, b2, (short)0, acc2, false, false);
        acc3 = __builtin_amdgcn_wmma_f32_16x16x4_f32(false, a, false, b3, (short)0, acc3, false, false);
    }

    const int mbase = (lane >> 4) * 8;   // VGPR r holds M = r + mbase
#pragma unroll
    for (int r = 0; r < 8; ++r) {
        float* hrow = h + (size_t)(row0 + mbase + r) * D + m;
        hrow[ 0] = acc0[r];
        hrow[16] = acc1[r];
        hrow[32] = acc2[r];
        hrow[48] = acc3[r];
    }
}

// --- 5) out[n,d] = deginv[n]*h[n,d] + b[d]  (self-loop term + bias) -----
__global__ __launch_bounds__(256) void k_init_out(const float* __restrict__ deginv,
                                                  const float* __restrict__ h,
                                                  const float* __restrict__ b,
                                                  float* __restrict__ out) {
    int idx = blockIdx.x * blockDim.x + threadIdx.x;
    if (idx < N_NODES * D) {
        int n = idx >> 6;
        int d = idx & (D - 1);
        out[idx] = deginv[n] * h[idx] + b[d];
    }
}

// --- 6) out[dst] += deginv[dst]*ew * h[src]  (one wave per edge) --------
__global__ __launch_bounds__(256) void k_edges(const long long* __restrict__ src,
                                               const long long* __restrict__ dst,
                                               const float* __restrict__ ew,
                                               const float* __restrict__ deginv,
                                               const float* __restrict__ h,
                                               float* __restrict__ out) {
    const int lane = threadIdx.x & 31;
    // force wave index scalar so per-edge metadata becomes SMEM loads
    const int w = __builtin_amdgcn_readfirstlane((int)(threadIdx.x >> 5));
    const long long e = (long long)blockIdx.x * 8 + w;
    if (e >= N_EDGES) return;

    const long long s = src[e];
    const long long t = dst[e];
    const float norm  = deginv[t] * ew[e];

    v2f hv = *(const v2f*)(h + s * D + lane * 2);
    float* op = out + t * D + lane * 2;
    unsafeAtomicAdd(op + 0, norm * hv.x);
    unsafeAtomicAdd(op + 1, norm * hv.y);
}

extern "C" void kernel_launch(void* const* d_in, const int* in_sizes, int n_in,
                              void* d_out, int out_size, void* d_ws, size_t ws_size,
                              hipStream_t stream) {
    const float*     x          = (const float*)d_in[0];
    const long long* edge_index = (const long long*)d_in[1];
    const float*     ew         = (const float*)d_in[2];
    const float*     W          = (const float*)d_in[3];
    const float*     b          = (const float*)d_in[4];
    float*           out        = (float*)d_out;

    const long long* src = edge_index;            // edge_index[0, :]
    const long long* dst = edge_index + N_EDGES;  // edge_index[1, :]

    char*  ws  = (char*)d_ws;
    float* deg = (float*)ws;                      // N_NODES f32 (becomes deginv)
    float* h   = (float*)(ws + (512u << 10));     // N_NODES*64 f32 (25.6 MB)

    k_init_deg  <<<(N_NODES + 255) / 256, 256, 0, stream>>>(deg);
    k_accum_deg <<<(N_EDGES + 255) / 256, 256, 0, stream>>>(dst, ew, deg);
    k_invert_deg<<<(N_NODES + 255) / 256, 256, 0, stream>>>(deg);
    k_gemm      <<<(N_NODES / 16 + 7) / 8, 256, 0, stream>>>(x, W, h);
    k_init_out  <<<(N_NODES * D + 255) / 256, 256, 0, stream>>>(deg, h, b, out);
    k_edges     <<<N_EDGES / 8, 256, 0, stream>>>(src, dst, ew, deg, h, out);
}